// IIRFilter_12541304504598
// MI455X (gfx1250) — compile-verified
//
#include <hip/hip_runtime.h>

// ---------------------------------------------------------------------------
// filtfilt (order-8 bandpass IIR, 17 taps) on x:(512,32768) f32 for MI455X.
//
// Block state-space over 16-step time blocks and 16-row batches:
//   Y(16x16) = T(16x16)*F(16x16) + U(16x16)*C(16x16)
// via V_WMMA_F32_16X16X4_F32.  The U*C product (4 dependent WMMAs) is the
// only true recurrence; T*F runs off the critical path in a second
// accumulator.  Each wave processes TWO independent 16-row groups of the
// same time chunk so two recurrence chains interleave in one instruction
// stream.  Poles |p|=0.8 => 512-sample warm-up decouples time chunks
// (0.8^512 ~ 1e-50), giving 128 independent wave-tasks.
// ---------------------------------------------------------------------------

#define TLEN   32768
#define KTAPS  17
#define NSTATE 16
#define NCHUNK 8
#define CHUNK  (TLEN / NCHUNK)   // 4096
#define WARM   512               // warm-up samples (multiple of 16)

typedef float v2f __attribute__((ext_vector_type(2)));
typedef float v8f __attribute__((ext_vector_type(8)));

// ws float layout: [0..16]=bn  [32..47]=an  [64..319]=T[m][k]  [320..575]=U[m][s]
// F buffer (512*32768 f32) at byte offset 4096.

// ---------------------------------------------------------------------------
__global__ void setup_kernel(const float* __restrict__ b,
                             const float* __restrict__ a,
                             float* __restrict__ coef) {
    if (threadIdx.x != 0 || blockIdx.x != 0) return;
    float a0 = a[0];
    float bn[KTAPS], an[NSTATE];
    for (int k = 0; k < KTAPS; ++k) { bn[k] = b[k] / a0; coef[k] = bn[k]; }
    for (int j = 0; j < NSTATE; ++j) { an[j] = a[j + 1] / a0; coef[32 + j] = an[j]; }

    // Impulse response of the recursive part within one block.
    float h[16];
    h[0] = 1.f;
    for (int m = 1; m < 16; ++m) {
        float s = 0.f;
        for (int j = 1; j <= m; ++j) s += an[j - 1] * h[m - j];
        h[m] = -s;
    }
    float* Tm = coef + 64;
    float* Um = coef + 320;
    for (int m = 0; m < 16; ++m)
        for (int k = 0; k < 16; ++k)
            Tm[m * 16 + k] = (m >= k) ? h[m - k] : 0.f;

    // U[:,s] = zero-input response to initial condition y[-1-s] = 1.
    for (int s = 0; s < 16; ++s) {
        float g[16];
        for (int n = 0; n < 16; ++n) {
            float acc = 0.f;
            for (int j = 1; j <= NSTATE; ++j) {
                int i = n - j;
                float v = (i >= 0) ? g[i] : ((i == -1 - s) ? 1.f : 0.f);
                acc -= an[j - 1] * v;
            }
            g[n] = acc;
        }
        for (int n = 0; n < 16; ++n) Um[n * 16 + s] = g[n];
    }
}

// ---------------------------------------------------------------------------
// FIR part, forward:  f[n] = sum_k bn[k] * x[n-k], zero initial conditions.
__global__ void fir_fwd_kernel(const float* __restrict__ x,
                               float* __restrict__ f,
                               const float* __restrict__ coef, long total) {
    long idx = (long)blockIdx.x * blockDim.x + threadIdx.x;
    if (idx >= total) return;
    int t = (int)(idx & (TLEN - 1));
    long rowOff = idx - t;
    float acc = 0.f;
#pragma unroll
    for (int k = 0; k < KTAPS; ++k)
        if (t - k >= 0) acc += coef[k] * x[rowOff + t - k];
    f[idx] = acc;
}

// FIR on time-reversed y1:  f2[n] = sum_k bn[k] * y1[T-1-n+k]  (k <= n).
__global__ void fir_rev_kernel(const float* __restrict__ y1,
                               float* __restrict__ f,
                               const float* __restrict__ coef, long total) {
    long idx = (long)blockIdx.x * blockDim.x + threadIdx.x;
    if (idx >= total) return;
    int n = (int)(idx & (TLEN - 1));
    long rowOff = idx - n;
    float acc = 0.f;
#pragma unroll
    for (int k = 0; k < KTAPS; ++k)
        if (k <= n) acc += coef[k] * y1[rowOff + (TLEN - 1 - n + k)];
    f[idx] = acc;
}

// ---------------------------------------------------------------------------
// WMMA block scan.  One wave per workgroup; a wave owns TWO 16-row groups
// over one time chunk.  REVERSE folds the final [::-1] into the stores.
template <bool REVERSE>
__launch_bounds__(32)
__global__ void iir_scan_kernel(const float* __restrict__ F,
                                float* __restrict__ Y,
                                const float* __restrict__ coef) {
    __shared__ __align__(16) float Flds[2][16][16]; // [grp][row n][local time]
    __shared__ __align__(16) float Clds[2][16][16]; // [grp][state s][row n]

    const int lane = threadIdx.x;
    const int n    = lane & 15;   // N index (row within group)
    const int half = lane >> 4;   // 0: lanes 0-15, 1: lanes 16-31

    const int task       = blockIdx.x;
    const int pairBase   = (task / NCHUNK) * 32;     // first row of group 0
    const int chunk      = task % NCHUNK;
    const int chunkStart = chunk * CHUNK;
    const int warmStart  = (chunkStart >= WARM) ? (chunkStart - WARM) : 0;
    const int chunkEnd   = chunkStart + CHUNK;

    // A fragments (16x4 f32, §7.12.2): lanes 0-15 hold K=kb,kb+1 (kb=4kc);
    // lanes 16-31 hold K=kb+2,kb+3.  Shared by both row groups.
    const float* Tm = coef + 64;
    const float* Um = coef + 320;
    v2f aT[4], aU[4];
#pragma unroll
    for (int kc = 0; kc < 4; ++kc) {
        int kb = 4 * kc + 2 * half;
        aT[kc].x = Tm[n * 16 + kb]; aT[kc].y = Tm[n * 16 + kb + 1];
        aU[kc].x = Um[n * 16 + kb]; aU[kc].y = Um[n * 16 + kb + 1];
    }

    // Zero both carries (each lane owns 8 of the 256 entries per group).
#pragma unroll
    for (int v = 0; v < 8; ++v) {
        Clds[0][v + 8 * half][n] = 0.f;
        Clds[1][v + 8 * half][n] = 0.f;
    }
    __syncthreads();

    const float* Frow0 = F + (long)(pairBase + n) * TLEN;
    const float* Frow1 = F + (long)(pairBase + 16 + n) * TLEN;
    float*       Yrow0 = Y + (long)(pairBase + n) * TLEN;
    float*       Yrow1 = Y + (long)(pairBase + 16 + n) * TLEN;

    // Per-lane output cursors (stride +/-16 floats per block => b128 stores
    // with immediate offsets, reversed addressing folded in).
    const int outOff = REVERSE ? (TLEN - 8 - chunkStart - half * 8)
                               : (chunkStart + half * 8);
    float* yp0 = Yrow0 + outOff;
    float* yp1 = Yrow1 + outOff;
    const int ystep = REVERSE ? -16 : 16;

    // One 16-step block for both row groups.  The U*C chains (4 dependent
    // WMMAs each) are the recurrence-critical path; the two groups' chains
    // interleave, and T*F runs in independent accumulators.
    auto block_step = [&](int t, bool doStore) {
        // Stage F tiles: 128-bit coalesced loads, 8 contiguous floats/lane.
        const float4 f00 = *(const float4*)(Frow0 + t + half * 8);
        const float4 f01 = *(const float4*)(Frow0 + t + half * 8 + 4);
        const float4 f10 = *(const float4*)(Frow1 + t + half * 8);
        const float4 f11 = *(const float4*)(Frow1 + t + half * 8 + 4);
        *(float4*)&Flds[0][n][half * 8]     = f00;
        *(float4*)&Flds[0][n][half * 8 + 4] = f01;
        *(float4*)&Flds[1][n][half * 8]     = f10;
        *(float4*)&Flds[1][n][half * 8 + 4] = f11;
        if (t + 16 < chunkEnd) {
            __builtin_prefetch(Frow0 + t + 16 + half * 8, 0, 1); // global_prefetch_b8
            __builtin_prefetch(Frow1 + t + 16 + half * 8, 0, 1);
        }
        __syncthreads();

        v8f accT0 = {0.f,0.f,0.f,0.f,0.f,0.f,0.f,0.f}, accU0 = accT0;
        v8f accT1 = accT0, accU1 = accT0;
#pragma unroll
        for (int kc = 0; kc < 4; ++kc) {          // off-critical-path: T * F
            v2f bf0, bf1;
            bf0.x = Flds[0][n][4 * kc + 2 * half];
            bf0.y = Flds[0][n][4 * kc + 2 * half + 1];
            bf1.x = Flds[1][n][4 * kc + 2 * half];
            bf1.y = Flds[1][n][4 * kc + 2 * half + 1];
            accT0 = __builtin_amdgcn_wmma_f32_16x16x4_f32(
                false, aT[kc], false, bf0, (short)0, accT0, false, false);
            accT1 = __builtin_amdgcn_wmma_f32_16x16x4_f32(
                false, aT[kc], false, bf1, (short)0, accT1, false, false);
        }
#pragma unroll
        for (int kc = 0; kc < 4; ++kc) {          // recurrence: U * C
            v2f bc0, bc1;
            bc0.x = Clds[0][4 * kc + 2 * half][n];
            bc0.y = Clds[0][4 * kc + 2 * half + 1][n];
            bc1.x = Clds[1][4 * kc + 2 * half][n];
            bc1.y = Clds[1][4 * kc + 2 * half + 1][n];
            accU0 = __builtin_amdgcn_wmma_f32_16x16x4_f32(
                false, aU[kc], false, bc0, (short)0, accU0, false, false);
            accU1 = __builtin_amdgcn_wmma_f32_16x16x4_f32(
                false, aU[kc], false, bc1, (short)0, accU1, false, false);
        }
        v8f acc0 = accT0 + accU0;
        v8f acc1 = accT1 + accU1;
        __syncthreads(); // all lanes done reading Clds/Flds before overwrite

        // Carry for next block: C[s][n] = Y[15-s][n].  D layout: lane holds
        // M = v (+8 for high lanes), N = n.
#pragma unroll
        for (int v = 0; v < 8; ++v) {
            int m = v + 8 * half;
            Clds[0][15 - m][n] = acc0[v];
            Clds[1][15 - m][n] = acc1[v];
        }

        if (doStore) {
            if (!REVERSE) {
                *(float4*)(yp0)     = (float4){acc0[0], acc0[1], acc0[2], acc0[3]};
                *(float4*)(yp0 + 4) = (float4){acc0[4], acc0[5], acc0[6], acc0[7]};
                *(float4*)(yp1)     = (float4){acc1[0], acc1[1], acc1[2], acc1[3]};
                *(float4*)(yp1 + 4) = (float4){acc1[4], acc1[5], acc1[6], acc1[7]};
            } else {
                *(float4*)(yp0)     = (float4){acc0[7], acc0[6], acc0[5], acc0[4]};
                *(float4*)(yp0 + 4) = (float4){acc0[3], acc0[2], acc0[1], acc0[0]};
                *(float4*)(yp1)     = (float4){acc1[7], acc1[6], acc1[5], acc1[4]};
                *(float4*)(yp1 + 4) = (float4){acc1[3], acc1[2], acc1[1], acc1[0]};
            }
            yp0 += ystep;
            yp1 += ystep;
        }
        __syncthreads(); // carry visible before next iteration reads it
    };

    // Warm-up (discard outputs), then main chunk (always store): no
    // per-iteration store predicate on the hot path.
    for (int t = warmStart; t < chunkStart; t += 16) block_step(t, false);
    for (int t = chunkStart; t < chunkEnd;   t += 16) block_step(t, true);
}

// ---------------------------------------------------------------------------
extern "C" void kernel_launch(void* const* d_in, const int* in_sizes, int n_in,
                              void* d_out, int out_size, void* d_ws, size_t ws_size,
                              hipStream_t stream) {
    const float* x = (const float*)d_in[0];
    const float* b = (const float*)d_in[1];
    const float* a = (const float*)d_in[2];
    float* out  = (float*)d_out;
    float* coef = (float*)d_ws;
    float* F    = (float*)((char*)d_ws + 4096);

    const long total = (long)in_sizes[0];          // 512*32768
    const int  B     = (int)(total / TLEN);
    const int  tasks = (B / 32) * NCHUNK;          // 128 wave-tasks, 2 groups each

    const int  blk  = 256;
    const long nblk = (total + blk - 1) / blk;

    setup_kernel<<<1, 1, 0, stream>>>(b, a, coef);
    // Pass 1: forward filter.
    fir_fwd_kernel<<<dim3((unsigned)nblk), blk, 0, stream>>>(x, F, coef, total);
    iir_scan_kernel<false><<<tasks, 32, 0, stream>>>(F, out, coef);
    // Pass 2: filter the time-reversed signal; store reversed back.
    fir_rev_kernel<<<dim3((unsigned)nblk), blk, 0, stream>>>(out, F, coef, total);
    iir_scan_kernel<true><<<tasks, 32, 0, stream>>>(F, out, coef);
}